// Involution_88699664597252
// MI455X (gfx1250) — compile-verified
//
#include <hip/hip_runtime.h>
#include <hip/hip_bf16.h>

// Involution (B=4, C=256, H=W=56, K=7, GROUPS=16) for MI455X / gfx1250.
// Stage 1: h = ReLU(BN(w1 @ x))        -- fp32 WMMA 16x16x4 GEMM, fused BN+ReLU
// Stage 2: wgt = w2 @ h + b2           -- fp32 WMMA 16x16x4 GEMM, fused bias
// Stage 3: out = 49-tap involution     -- coalesced VALU FMA kernel
//
// Problem is HBM-bound (~80 MB @ 23.3 TB/s), so fp32 WMMA (not bf16) keeps
// reference precision at zero cost.

#define Bn 4
#define Cn 256
#define Hn 56
#define Wn 56
#define HWn 3136          // 56*56 = 196 * 16
#define CMID 64           // C/4
#define KK 49
#define NGROUP 16
#define KKG 784           // 49*16
#define NTILES 196        // HW / 16

typedef float v2f __attribute__((ext_vector_type(2)));
typedef float v8f __attribute__((ext_vector_type(8)));

// ---------------------------------------------------------------------------
// Stage 1: h[b, 64, HW] = ReLU(BN(w1[64,256] @ x[b, 256, HW]))
// One wave32 computes one 16x16 output tile; K loop steps by 4 (WMMA f32).
// Tiles: Bn * (64/16) * NTILES = 4*4*196 = 3136 = 392 blocks * 8 waves exactly.
// ---------------------------------------------------------------------------
__global__ __launch_bounds__(256) void inv_gen1(
    const float* __restrict__ x, const float* __restrict__ w1,
    const float* __restrict__ gamma, const float* __restrict__ beta,
    const float* __restrict__ mean, const float* __restrict__ var,
    float* __restrict__ hbuf)
{
    const int lane = threadIdx.x & 31;
    const int wave = blockIdx.x * (blockDim.x >> 5) + (threadIdx.x >> 5);

    const int nt = wave % NTILES;       // pixel tile
    const int t  = wave / NTILES;
    const int mt = t & 3;               // out-channel tile (0..3)
    const int b  = t >> 2;              // batch

    const int m  = lane & 15;           // M index (A) / N index (B,D) per lane
    const int hi = lane >> 4;           // half-wave selector (K pair / M+8)

    const float* Arow = w1 + (mt * 16 + m) * Cn;            // row-major [64,256]
    const float* Bcol = x + b * (Cn * HWn) + nt * 16 + m;   // [256, HW]

    v8f acc = {};
#pragma unroll 4
    for (int k = 0; k < Cn; k += 4) {
        const int ka = k + 2 * hi;
        v2f a, bv;
        a.x  = Arow[ka];
        a.y  = Arow[ka + 1];
        bv.x = Bcol[(size_t)ka * HWn];
        bv.y = Bcol[(size_t)(ka + 1) * HWn];
        acc = __builtin_amdgcn_wmma_f32_16x16x4_f32(
            false, a, false, bv, (short)0, acc, false, false);
    }

    // D layout: VGPR r -> M = mt*16 + 8*hi + r, N = nt*16 + (lane&15)
    const int n = nt * 16 + m;
    float* Hb = hbuf + (size_t)b * (CMID * HWn) + n;
#pragma unroll
    for (int r = 0; r < 8; ++r) {
        const int mo = mt * 16 + 8 * hi + r;
        const float s  = gamma[mo] * rsqrtf(var[mo] + 1e-5f);
        const float sh = beta[mo] - mean[mo] * s;
        float v = fmaf(acc[r], s, sh);
        Hb[(size_t)mo * HWn] = fmaxf(v, 0.0f);
    }
}

// ---------------------------------------------------------------------------
// Stage 2: wgt[b, 784, HW] = w2[784,64] @ h[b, 64, HW] + b2
// Tiles: Bn * (784/16) * NTILES = 4*49*196 = 38416 = 4802 blocks * 8 waves.
// ---------------------------------------------------------------------------
__global__ __launch_bounds__(256) void inv_gen2(
    const float* __restrict__ hbuf, const float* __restrict__ w2,
    const float* __restrict__ b2, float* __restrict__ wbuf)
{
    const int lane = threadIdx.x & 31;
    const int wave = blockIdx.x * (blockDim.x >> 5) + (threadIdx.x >> 5);

    const int nt = wave % NTILES;
    const int t  = wave / NTILES;
    const int mt = t % 49;
    const int b  = t / 49;

    const int m  = lane & 15;
    const int hi = lane >> 4;

    const float* Arow = w2 + (mt * 16 + m) * CMID;               // [784,64]
    const float* Bcol = hbuf + (size_t)b * (CMID * HWn) + nt * 16 + m;

    v8f acc = {};
#pragma unroll
    for (int k = 0; k < CMID; k += 4) {
        const int ka = k + 2 * hi;
        v2f a, bv;
        a.x  = Arow[ka];
        a.y  = Arow[ka + 1];
        bv.x = Bcol[(size_t)ka * HWn];
        bv.y = Bcol[(size_t)(ka + 1) * HWn];
        acc = __builtin_amdgcn_wmma_f32_16x16x4_f32(
            false, a, false, bv, (short)0, acc, false, false);
    }

    const int n = nt * 16 + m;
    float* Wb = wbuf + (size_t)b * (KKG * HWn) + n;
#pragma unroll
    for (int r = 0; r < 8; ++r) {
        const int mo = mt * 16 + 8 * hi + r;
        Wb[(size_t)mo * HWn] = acc[r] + b2[mo];
    }
}

// ---------------------------------------------------------------------------
// Stage 3: out[b,c,y,x] = sum_{ki,kj} x[b,c,y+ki-3,x+kj-3] * wgt[b, g*49+ki*7+kj, y, x]
// One thread per output element; weight + x reads are coalesced along x (w).
// Total = 4*256*3136 = 3,211,264 = 12544 blocks * 256 threads exactly.
// ---------------------------------------------------------------------------
__global__ __launch_bounds__(256) void inv_apply(
    const float* __restrict__ x, const float* __restrict__ wbuf,
    float* __restrict__ out)
{
    const int idx = blockIdx.x * 256 + threadIdx.x;
    const int hw = idx % HWn;
    const int t  = idx / HWn;
    const int ch = t % Cn;
    const int b  = t / Cn;
    const int y  = hw / Wn;
    const int xw = hw % Wn;
    const int g  = ch >> 4;     // 16 channels per group

    const float* wb = wbuf + ((size_t)b * KKG + (size_t)g * KK) * HWn + hw;
    const float* xb = x + ((size_t)b * Cn + ch) * HWn;

    float acc = 0.0f;
#pragma unroll
    for (int ki = 0; ki < 7; ++ki) {
        const int iy = y + ki - 3;
        if ((unsigned)iy >= (unsigned)Hn) continue;
#pragma unroll
        for (int kj = 0; kj < 7; ++kj) {
            const int ix = xw + kj - 3;
            if ((unsigned)ix < (unsigned)Wn) {
                acc = fmaf(xb[iy * Wn + ix],
                           wb[(size_t)(ki * 7 + kj) * HWn], acc);
            }
        }
    }
    out[idx] = acc;
}

extern "C" void kernel_launch(void* const* d_in, const int* in_sizes, int n_in,
                              void* d_out, int out_size, void* d_ws, size_t ws_size,
                              hipStream_t stream)
{
    const float* x     = (const float*)d_in[0];
    const float* w1    = (const float*)d_in[1];
    const float* gamma = (const float*)d_in[2];
    const float* beta  = (const float*)d_in[3];
    const float* mean  = (const float*)d_in[4];
    const float* var   = (const float*)d_in[5];
    const float* w2    = (const float*)d_in[6];
    const float* b2    = (const float*)d_in[7];
    float* out = (float*)d_out;

    // workspace: h = [4, 64, 3136] (3.2 MB), wgt = [4, 784, 3136] (39.3 MB)
    float* hbuf = (float*)d_ws;
    float* wbuf = hbuf + (size_t)Bn * CMID * HWn;

    inv_gen1<<<392, 256, 0, stream>>>(x, w1, gamma, beta, mean, var, hbuf);
    inv_gen2<<<4802, 256, 0, stream>>>(hbuf, w2, b2, wbuf);
    inv_apply<<<12544, 256, 0, stream>>>(x, wbuf, out);
}